// AOSPredictionLayer_53283364274772
// MI455X (gfx1250) — compile-verified
//
#include <hip/hip_runtime.h>
#include <hip/hip_bf16.h>

// Problem constants (match reference)
#define BB 16384
#define NN 20
#define DD 64
#define RR 3
#define KK 64
#define TWOD 128

typedef __attribute__((ext_vector_type(2))) float v2f;
typedef __attribute__((ext_vector_type(8))) float v8f;

#define LDSW_STRIDE 130   // 128 + 2 pad -> bank-conflict-free ds_load_b64

// ---------------------------------------------------------------------------
// Kernel 1: ui_all[b][r][k] = concat(u_emb,i_emb)[b,:] @ w_uir[r] + r_vec[r]
// grid = B/16, block = 96 (3 waves, wave id == relation)
// ---------------------------------------------------------------------------
__global__ __launch_bounds__(96)
void ui_gemm_kernel(const float* __restrict__ u_emb,
                    const float* __restrict__ i_emb,
                    const float* __restrict__ w_uir,
                    const float* __restrict__ r_vec,
                    float* __restrict__ ui_all) {
    const int r    = threadIdx.x >> 5;   // relation == wave id (0..2)
    const int lane = threadIdx.x & 31;
    const int half = lane >> 4;
    const int lrow = lane & 15;
    const int row  = blockIdx.x * 16 + lrow;   // batch index for A fragment

    v8f acc[4] = {};
    const float* wr = w_uir + r * TWOD * KK;

    #pragma unroll
    for (int ks = 0; ks < 32; ++ks) {
        const int k0 = ks * 4 + half * 2;       // even, never straddles 64
        const float* p = (k0 < DD) ? (u_emb + row * DD + k0)
                                   : (i_emb + row * DD + (k0 - DD));
        v2f a; a.x = p[0]; a.y = p[1];
        #pragma unroll
        for (int ct = 0; ct < 4; ++ct) {
            const int c = ct * 16 + lrow;
            v2f b;
            b.x = wr[k0 * KK + c];
            b.y = wr[(k0 + 1) * KK + c];
            acc[ct] = __builtin_amdgcn_wmma_f32_16x16x4_f32(
                false, a, false, b, (short)0, acc[ct], false, false);
        }
    }

    // bias + scattered store: element (m, c) lives in acc[ct][v], m = v + 8*half
    #pragma unroll
    for (int ct = 0; ct < 4; ++ct) {
        const int c  = ct * 16 + lrow;
        const float rv = r_vec[r * KK + c];
        #pragma unroll
        for (int v = 0; v < 8; ++v) {
            const int m = v + half * 8;
            const int b = blockIdx.x * 16 + m;
            ui_all[(b * RR + r) * KK + c] = acc[ct][v] + rv;
        }
    }
}

// ---------------------------------------------------------------------------
// Kernel 2: per 16-row tile of ao rows (row = b*N + n):
//   D[m,:] = sum_r mask_r(m) * (ao_in[m,:] @ w_aor[r])   (relation-masked A,
//            accumulated into ONE chained WMMA accumulator)
//   pred[m] = sum_k D[m,k] * ui_all[b(m), s(m), k]
// grid = (B*N)/(16*8), block = 256 (8 waves, one 16-row tile per wave)
// ---------------------------------------------------------------------------
__global__ __launch_bounds__(256)
void ao_pred_kernel(const float* __restrict__ a_emb,
                    const float* __restrict__ o_emb,
                    const int*   __restrict__ s,
                    const float* __restrict__ w_aor,
                    const float* __restrict__ ui_all,
                    float* __restrict__ pred) {
    extern __shared__ float smem[];
    float* ldsW  = smem;                                   // 3*64*130 floats
    float* ldsUI = smem + RR * KK * LDSW_STRIDE;           // 8 * 16*64 floats
    float* ldsR  = ldsUI + 8 * 16 * KK;                    // 8 * 16*17 floats

    const int tid  = threadIdx.x;
    const int wave = tid >> 5;
    const int lane = tid & 31;
    const int half = lane >> 4;
    const int lrow = lane & 15;

    // ---- stage w_aor into LDS, transposed to [r][c][k] with pad ----
    #pragma unroll 4
    for (int i = 0; i < 96; ++i) {
        const int idx = tid + i * 256;          // 0 .. 24575
        const int r   = idx >> 13;              // / (128*64)
        const int rem = idx & 8191;
        const int k   = rem >> 6;
        const int c   = rem & 63;
        ldsW[(r * KK + c) * LDSW_STRIDE + k] = w_aor[idx];
    }

    const int tile = blockIdx.x * 8 + wave;
    const int row  = tile * 16 + lrow;          // this lane's A-matrix row

    // ---- stage ui_sel[m][k] = ui_all[b(m), s(m), k] for the 16 tile rows ----
    float* myUI = ldsUI + wave * (16 * KK);
    #pragma unroll
    for (int m = 0; m < 16; ++m) {
        const int rm   = tile * 16 + m;
        const int bm   = rm / NN;
        const int relm = s[rm];
        const float* src = ui_all + (bm * RR + relm) * KK;
        myUI[m * KK + lane]      = src[lane];
        myUI[m * KK + lane + 32] = src[lane + 32];
    }

    // ---- preload this lane's A values (ao_in row) into VGPRs ----
    float areg[64];
    const float* pa = a_emb + (size_t)row * DD;
    const float* po = o_emb + (size_t)row * DD;
    #pragma unroll
    for (int ks = 0; ks < 32; ++ks) {
        const int k0 = ks * 4 + half * 2;       // even, never straddles 64
        const float* p = (k0 < DD) ? (pa + k0) : (po + k0 - DD);
        areg[2 * ks]     = p[0];
        areg[2 * ks + 1] = p[1];
    }

    const int myrel = s[row];

    __syncthreads();   // ldsW + ldsUI visible

    // ---- relation-masked chained WMMA: effective K = 3 * 128 ----
    v8f acc[4] = {};
    for (int r = 0; r < RR; ++r) {
        const float maskf = (myrel == r) ? 1.0f : 0.0f;
        const float* wrow = ldsW + r * KK * LDSW_STRIDE;
        #pragma unroll
        for (int ks = 0; ks < 32; ++ks) {
            const int k0 = ks * 4 + half * 2;
            v2f a;
            a.x = areg[2 * ks]     * maskf;
            a.y = areg[2 * ks + 1] * maskf;
            #pragma unroll
            for (int ct = 0; ct < 4; ++ct) {
                const int c = ct * 16 + lrow;
                const v2f b = *(const v2f*)(wrow + c * LDSW_STRIDE + k0); // ds_load_b64
                acc[ct] = __builtin_amdgcn_wmma_f32_16x16x4_f32(
                    false, a, false, b, (short)0, acc[ct], false, false);
            }
        }
    }

    // ---- epilogue: pred[m] = sum_c D[m,c] * ui_sel[m,c] ----
    // D element (m, c): acc[ct][v], m = v + 8*half, c = ct*16 + lrow
    float part[8];
    #pragma unroll
    for (int v = 0; v < 8; ++v) {
        const int m = v + half * 8;
        float sum = 0.f;
        #pragma unroll
        for (int ct = 0; ct < 4; ++ct)
            sum += acc[ct][v] * myUI[m * KK + ct * 16 + lrow];
        part[v] = sum;
    }

    float* myR = ldsR + wave * (16 * 17);
    #pragma unroll
    for (int v = 0; v < 8; ++v) {
        const int m = v + half * 8;
        myR[m * 17 + lrow] = part[v];           // padded: conflict-free readback
    }
    __syncthreads();

    if (lane < 16) {
        float sum = 0.f;
        #pragma unroll
        for (int j = 0; j < 16; ++j) sum += myR[lane * 17 + j];
        pred[tile * 16 + lane] = sum;
    }
}

// ---------------------------------------------------------------------------
extern "C" void kernel_launch(void* const* d_in, const int* in_sizes, int n_in,
                              void* d_out, int out_size, void* d_ws, size_t ws_size,
                              hipStream_t stream) {
    const float* u_emb = (const float*)d_in[0];
    const float* i_emb = (const float*)d_in[1];
    const float* a_emb = (const float*)d_in[2];
    const float* o_emb = (const float*)d_in[3];
    const int*   s     = (const int*)  d_in[4];
    const float* w_uir = (const float*)d_in[5];
    const float* w_aor = (const float*)d_in[6];
    const float* r_vec = (const float*)d_in[7];
    float* pred   = (float*)d_out;
    float* ui_all = (float*)d_ws;    // B * R * K floats = 12.6 MB scratch

    // Kernel 1: ui GEMM (+bias) into scratch
    ui_gemm_kernel<<<BB / 16, 96, 0, stream>>>(u_emb, i_emb, w_uir, r_vec, ui_all);

    // Kernel 2: masked ao GEMM + fused dot-product reduction
    const size_t smem_bytes =
        (size_t)(RR * KK * LDSW_STRIDE + 8 * 16 * KK + 8 * 16 * 17) * sizeof(float);
    const int n_tiles = (BB * NN) / 16;       // 20480
    ao_pred_kernel<<<n_tiles / 8, 256, smem_bytes, stream>>>(
        a_emb, o_emb, s, w_aor, ui_all, pred);
}